// CLUBMean_48704929137027
// MI455X (gfx1250) — compile-verified
//
#include <hip/hip_runtime.h>
#include <hip/hip_bf16.h>

// ---------------------------------------------------------------------------
// CLUB-mean estimator, restructured:
//   result = sum_d [ (1/N) * sum_i mu[i][d]*y[i][d]
//                    - (sum_i mu[i][d]/N) * (sum_j y[j][d]/N) ]
// (the E[y^2] terms of positive/negative cancel exactly under the mean over i,
//  removing the naive O(N^2 D) negative term entirely).
// Cost model (MI455X): pooling reads 201 MB -> ~9us @ 23.3 TB/s (HBM bound).
// GEMMs are 0.8 GFLOP, run on V_WMMA_F32_16X16X4_F32 (full f32 precision),
// LDS double-buffered so WMMA never waits on an L2 round trip.
// ---------------------------------------------------------------------------

typedef float v2f __attribute__((ext_vector_type(2)));
typedef float v8f __attribute__((ext_vector_type(8)));

#define Nn   1024
#define XC   512
#define YC   256
#define HID  512

// GEMM tiling
#define BM   128          // block tile M (8 waves x 16)
#define BN   64           // block tile N (4 wmma subtiles of 16)
#define KC   32           // K chunk staged in LDS
#define PADA 36           // KC + 4  : A LDS row stride (bank-conflict-free frag reads)
#define PADB 68           // BN + 4  : B LDS row stride

// --- 1) spatial mean: wave-cooperative, fully coalesced 512B requests ------
// Lanes 0-15 reduce output o, lanes 16-31 reduce output o+1; 4 passes/wave.
__global__ __launch_bounds__(256)
void pool64_kernel(const float* __restrict__ in, float* __restrict__ out, int nOut) {
    const int wave = (blockIdx.x * blockDim.x + threadIdx.x) >> 5;
    const int lane = threadIdx.x & 31;
    const int base = wave * 8;
#pragma unroll
    for (int p = 0; p < 4; ++p) {
        const int o = base + p * 2 + (lane >> 4);
        if (o < nOut) {
            const float4* ptr = (const float4*)(in + (size_t)o * 64);
            float4 v = ptr[lane & 15];
            float s = (v.x + v.y) + (v.z + v.w);
            // xor masks 1,2,4,8 stay within each 16-lane half of the wave32
            s += __shfl_xor(s, 1, 32);
            s += __shfl_xor(s, 2, 32);
            s += __shfl_xor(s, 4, 32);
            s += __shfl_xor(s, 8, 32);
            if ((lane & 15) == 0) out[o] = s * (1.0f / 64.0f);
        }
    }
}

// --- 2) f32 WMMA GEMM, LDS double-buffered -------------------------------
// C = act(A[M,K] @ B[K,Ncol] + bias). Block: 256 thr = 8 waves, tile 128x64.
// Wave w computes rows [w*16, w*16+16) of the block tile, all 64 cols.
// A-frag (ISA 16x4 f32): lanes 0-15 -> M=lane, K=k,k+1 ; lanes 16-31 -> K=k+2,k+3.
__global__ __launch_bounds__(256)
void gemm_wmma_f32(const float* __restrict__ A, const float* __restrict__ Bw,
                   const float* __restrict__ bias, float* __restrict__ C,
                   int M, int K, int Ncol, int do_relu)
{
    __shared__ float As[2][BM * PADA];   // 2*128*36*4 = 36864 B
    __shared__ float Bs[2][KC * PADB];   // 2* 32*68*4 = 17408 B

    const int tid  = threadIdx.x;
    const int lane = tid & 31;
    const int w    = tid >> 5;
    const int tilesN = Ncol / BN;
    const int m0 = (blockIdx.x / tilesN) * BM;
    const int n0 = (blockIdx.x % tilesN) * BN;

    const int half = lane >> 4;          // 0: lanes 0-15, 1: lanes 16-31
    const int l16  = lane & 15;
    const int rowL = w * 16 + l16;       // A LDS row for this lane

    // per-chunk staging registers (issued early, consumed after compute)
    float4 ra[4], rb[2];

    auto loadA = [&](int k0) {
#pragma unroll
        for (int i = 0; i < 4; ++i) {
            int q = i * 256 + tid;       // BM*KC/4 = 1024 float4
            int row = q >> 3;            // 8 float4 per 32-float row
            int c4  = q & 7;
            ra[i] = *(const float4*)(A + (size_t)(m0 + row) * K + k0 + c4 * 4);
        }
    };
    auto loadB = [&](int k0) {
#pragma unroll
        for (int i = 0; i < 2; ++i) {
            int q = i * 256 + tid;       // KC*BN/4 = 512 float4
            int row = q >> 4;            // 16 float4 per 64-float row
            int c4  = q & 15;
            rb[i] = *(const float4*)(Bw + (size_t)(k0 + row) * Ncol + n0 + c4 * 4);
        }
    };
    auto storeAB = [&](int buf) {
#pragma unroll
        for (int i = 0; i < 4; ++i) {
            int q = i * 256 + tid;
            int row = q >> 3, c4 = q & 7;
            *(float4*)&As[buf][row * PADA + c4 * 4] = ra[i];
        }
#pragma unroll
        for (int i = 0; i < 2; ++i) {
            int q = i * 256 + tid;
            int row = q >> 4, c4 = q & 15;
            *(float4*)&Bs[buf][row * PADB + c4 * 4] = rb[i];
        }
    };

    v8f acc[4] = {};
    auto compute = [&](int buf) {
#pragma unroll
        for (int kk = 0; kk < KC; kk += 4) {
            v2f a = *(const v2f*)&As[buf][rowL * PADA + kk + 2 * half];
#pragma unroll
            for (int s = 0; s < 4; ++s) {
                v2f b;
                b.x = Bs[buf][(kk + 2 * half)     * PADB + s * 16 + l16];
                b.y = Bs[buf][(kk + 2 * half + 1) * PADB + s * 16 + l16];
                acc[s] = __builtin_amdgcn_wmma_f32_16x16x4_f32(
                    false, a, false, b, (short)0, acc[s], false, false);
            }
        }
    };

    // prologue
    loadA(0); loadB(0);
    storeAB(0);
    __syncthreads();

    const int nChunks = K / KC;
    for (int c = 0; c < nChunks; ++c) {
        const int cur = c & 1;
        if (c + 1 < nChunks) { loadA((c + 1) * KC); loadB((c + 1) * KC); }
        compute(cur);                            // ds_load-fed WMMAs
        if (c + 1 < nChunks) storeAB(cur ^ 1);   // global latency hidden by compute
        __syncthreads();
    }

    // epilogue: +bias, optional ReLU, scatter per C/D layout
    // (acc VGPR r: lanes 0-15 -> row r, lanes 16-31 -> row r+8)
#pragma unroll
    for (int s = 0; s < 4; ++s) {
        const int col = n0 + s * 16 + l16;
        const float bv = bias[col];
#pragma unroll
        for (int r = 0; r < 8; ++r) {
            const int row = m0 + w * 16 + r + 8 * half;
            float v = acc[s][r] + bv;
            if (do_relu) v = fmaxf(v, 0.0f);
            C[(size_t)row * Ncol + col] = v;
        }
    }
}

// --- 3) per-column stats: dot(mu,y), sum(mu), sum(y) over rows ------------
__global__ __launch_bounds__(256)
void col_stats_kernel(const float* __restrict__ mu, const float* __restrict__ yv,
                      float* __restrict__ dotc, float* __restrict__ smc,
                      float* __restrict__ syc, int rows, int D)
{
    const int d = blockIdx.x;
    const int t = threadIdx.x;
    float dot = 0.f, sm = 0.f, sy = 0.f;
    for (int i = t; i < rows; i += blockDim.x) {
        float m = mu[(size_t)i * D + d];
        float y = yv[(size_t)i * D + d];
        dot += m * y; sm += m; sy += y;
    }
    __shared__ float sd[256], ssm[256], ssy[256];
    sd[t] = dot; ssm[t] = sm; ssy[t] = sy;
    __syncthreads();
    for (int off = 128; off > 0; off >>= 1) {
        if (t < off) { sd[t] += sd[t + off]; ssm[t] += ssm[t + off]; ssy[t] += ssy[t + off]; }
        __syncthreads();
    }
    if (t == 0) { dotc[d] = sd[0]; smc[d] = ssm[0]; syc[d] = ssy[0]; }
}

// --- 4) final scalar ------------------------------------------------------
__global__ __launch_bounds__(256)
void finalize_kernel(const float* __restrict__ dotc, const float* __restrict__ smc,
                     const float* __restrict__ syc, float* __restrict__ out,
                     int D, float invN)
{
    __shared__ float s[256];
    const int t = threadIdx.x;
    float v = 0.f;
    if (t < D) v = dotc[t] * invN - (smc[t] * invN) * (syc[t] * invN);
    s[t] = v;
    __syncthreads();
    for (int off = 128; off > 0; off >>= 1) {
        if (t < off) s[t] += s[t + off];
        __syncthreads();
    }
    if (t == 0) out[0] = s[0];
}

extern "C" void kernel_launch(void* const* d_in, const int* in_sizes, int n_in,
                              void* d_out, int out_size, void* d_ws, size_t ws_size,
                              hipStream_t stream) {
    const float* x_samples = (const float*)d_in[0];   // (1024,512,8,8)
    const float* y_samples = (const float*)d_in[1];   // (1024,256,8,8)
    const float* W1 = (const float*)d_in[2];          // (512,512)
    const float* b1 = (const float*)d_in[3];          // (512)
    const float* W2 = (const float*)d_in[4];          // (512,256)
    const float* b2 = (const float*)d_in[5];          // (256)
    float* out = (float*)d_out;

    float* ws = (float*)d_ws;
    float* x_vec = ws;                         // 1024*512
    float* y_vec = x_vec + (size_t)Nn * XC;    // 1024*256
    float* h     = y_vec + (size_t)Nn * YC;    // 1024*512
    float* mu    = h     + (size_t)Nn * HID;   // 1024*256
    float* dotc  = mu    + (size_t)Nn * YC;    // 256
    float* smc   = dotc + YC;                  // 256
    float* syc   = smc  + YC;                  // 256

    // 1) pooling (HBM-bound: ~201 MB). 1 wave -> 8 outputs, 512B/request.
    {
        int nOutX = Nn * XC;                   // 524288 -> 8192 blocks
        int nOutY = Nn * YC;                   // 262144 -> 4096 blocks
        pool64_kernel<<<nOutX / 64, 256, 0, stream>>>(x_samples, x_vec, nOutX);
        pool64_kernel<<<nOutY / 64, 256, 0, stream>>>(y_samples, y_vec, nOutY);
    }

    // 2) h = relu(x_vec @ W1 + b1) : M=1024, K=512, Ncol=512 -> 8x8 tiles
    gemm_wmma_f32<<<(Nn / BM) * (HID / BN), 256, 0, stream>>>(x_vec, W1, b1, h,
                                                              Nn, XC, HID, 1);

    // 3) mu = h @ W2 + b2 : M=1024, K=512, Ncol=256 -> 8x4 tiles
    gemm_wmma_f32<<<(Nn / BM) * (YC / BN), 256, 0, stream>>>(h, W2, b2, mu,
                                                             Nn, HID, YC, 0);

    // 4) column stats + 5) final scalar (deterministic tree reductions)
    col_stats_kernel<<<YC, 256, 0, stream>>>(mu, y_vec, dotc, smc, syc, Nn, YC);
    finalize_kernel<<<1, 256, 0, stream>>>(dotc, smc, syc, out, YC, 1.0f / (float)Nn);
}